// SparseEdgeEmbedding_46420006535593
// MI455X (gfx1250) — compile-verified
//
#include <hip/hip_runtime.h>
#include <hip/hip_bf16.h>

typedef __attribute__((ext_vector_type(2))) float v2f;
typedef __attribute__((ext_vector_type(8))) float v8f;

constexpr int NPTS   = 12000;
constexpr int TK     = 32;    // top-k
constexpr int NSIG   = 64;    // sigma count
constexpr int TILE   = 16;
constexpr int NTILES = NPTS / TILE;  // 750, exact
constexpr int NWAVES = 4;            // waves per block, each sweeps 1/4 of columns

// 4 wave32 per 16-row tile; WMMA f32 16x16x4 computes 16x16 squared-distance
// tiles (sqrt provably unnecessary: ordering and RBF both consume d^2):
//   A row m (K=4): [-2x_m, -2y_m, -2z_m, |x_m|^2],  B col n: [x_n, y_n, z_n, 1]
//   => C[m][n] = |x_m|^2 - 2 x_m . x_n ;  d2 = C + |x_n|^2
__global__ __launch_bounds__(128)
void SparseEdgeEmbedding_knn_rbf(const float* __restrict__ coords,
                                 const float* __restrict__ sigma,
                                 int*   __restrict__ idx_out,
                                 float* __restrict__ val_out)
{
    const int tid  = threadIdx.x;
    const int wv   = tid >> 5;
    const int lane = tid & 31;
    const int rt   = blockIdx.x;          // row tile index

    __shared__ float s_heap_d[NWAVES][16][TK];  // per-wave partial lists, DESC ([0]=max)
    __shared__ int   s_heap_i[NWAVES][16][TK];
    __shared__ int   s_cnt[NWAVES][16];
    __shared__ float s_thresh[NWAVES][16];
    __shared__ float s_stage[NWAVES][16][16];   // staged d^2 tile (slow path)
    __shared__ float s_inv[NSIG];               // 1/(2 sigma^2)
    __shared__ float s_fin_d[16][TK];           // merged final top-k, ASCENDING
    __shared__ int   s_fin_i[16][TK];

    if (lane < 16) { s_cnt[wv][lane] = 0; s_thresh[wv][lane] = 3.0e38f; }
    for (int s = tid; s < NSIG; s += 128) {
        float sg = sigma[s];
        s_inv[s] = 1.0f / (2.0f * sg * sg);
    }
    __syncthreads();

    // ---- A fragment (16x4 f32, per-ISA layout; identical in all 4 waves) ----
    // VGPR0: lanes0-15 -> K=0, lanes16-31 -> K=2 ; VGPR1: lanes0-15 -> K=1, lanes16-31 -> K=3
    const int m    = lane & 15;
    const int row  = rt * TILE + m;
    const float ax = coords[row * 3 + 0];
    const float ay = coords[row * 3 + 1];
    const float az = coords[row * 3 + 2];
    const float asq = ax * ax + ay * ay + az * az;
    v2f A;
    if (lane < 16) { A.x = -2.0f * ax; A.y = -2.0f * ay; }
    else           { A.x = -2.0f * az; A.y = asq;        }

    const int mbase = (lane < 16) ? 0 : 8;

    // VGPR-resident thresholds (reloaded from LDS only after a hit)
    float thv[8];
    #pragma unroll
    for (int v = 0; v < 8; ++v) thv[v] = 3.0e38f;

    // One tile of work: build B, WMMA, screen, (rare) heap insert.
    auto process_tile = [&](int t, float cx, float cy, float cz) {
        const float bsq = cx * cx + cy * cy + cz * cz;
        v2f B;                     // V0 = rows K0/K1, V1 = rows K2/K3
        if (lane < 16) { B.x = cx; B.y = cz;   }
        else           { B.x = cy; B.y = 1.0f; }

        v8f C = {};
        C = __builtin_amdgcn_wmma_f32_16x16x4_f32(
                false, A, false, B, (short)0, C, false, false);

        float d2v[8];
        #pragma unroll
        for (int v = 0; v < 8; ++v)
            d2v[v] = C[v] + bsq;               // unclamped d^2 (same ordering)

        // wave-wide threshold screen (fast path: reject whole 16x16 tile, 0 LDS)
        unsigned long long hit = 0ull;
        #pragma unroll
        for (int v = 0; v < 8; ++v)
            hit |= __ballot(d2v[v] < thv[v]);

        if (hit) {   // uniform within the wave; all LDS here is wave-private
            #pragma unroll
            for (int v = 0; v < 8; ++v)
                s_stage[wv][mbase + v][lane & 15] = fmaxf(d2v[v], 0.0f);
            // same-wave LDS is in-order (DScnt); just stop compiler reordering
            asm volatile("s_wait_dscnt 0x0" ::: "memory");

            if (lane < 16) {            // lane r owns row r's sorted list
                const int r  = lane;
                int   cnt = s_cnt[wv][r];
                float th  = s_thresh[wv][r];
                for (int j = 0; j < 16; ++j) {
                    const float d  = s_stage[wv][r][j];
                    const int   ci = t * TILE + j;
                    if (cnt < TK) {
                        int p = cnt;
                        while (p > 0 && s_heap_d[wv][r][p - 1] < d) {
                            s_heap_d[wv][r][p] = s_heap_d[wv][r][p - 1];
                            s_heap_i[wv][r][p] = s_heap_i[wv][r][p - 1];
                            --p;
                        }
                        s_heap_d[wv][r][p] = d; s_heap_i[wv][r][p] = ci;
                        if (++cnt == TK) th = s_heap_d[wv][r][0];
                    } else if (d < th) {
                        int p = 0;      // evict current max at [0], slide, insert
                        while (p < TK - 1 && s_heap_d[wv][r][p + 1] > d) {
                            s_heap_d[wv][r][p] = s_heap_d[wv][r][p + 1];
                            s_heap_i[wv][r][p] = s_heap_i[wv][r][p + 1];
                            ++p;
                        }
                        s_heap_d[wv][r][p] = d; s_heap_i[wv][r][p] = ci;
                        th = s_heap_d[wv][r][0];
                    }
                }
                s_cnt[wv][r] = cnt; s_thresh[wv][r] = th;
            }
            asm volatile("s_wait_dscnt 0x0" ::: "memory");
            #pragma unroll
            for (int v = 0; v < 8; ++v)        // refresh VGPR threshold cache
                thv[v] = s_thresh[wv][mbase + v];
        }
    };

    // ---- Column sweep over this wave's segment, distance-2 prefetch ----
    const int t0 = (wv * NTILES) / NWAVES;        // segments >= 187 tiles
    const int t1 = ((wv + 1) * NTILES) / NWAVES;

    const int cb = (lane & 15) * 3;
    float x0 = coords[(size_t)(t0 * TILE * 3) + cb + 0];
    float y0 = coords[(size_t)(t0 * TILE * 3) + cb + 1];
    float z0 = coords[(size_t)(t0 * TILE * 3) + cb + 2];
    float x1 = coords[(size_t)((t0 + 1) * TILE * 3) + cb + 0];
    float y1 = coords[(size_t)((t0 + 1) * TILE * 3) + cb + 1];
    float z1 = coords[(size_t)((t0 + 1) * TILE * 3) + cb + 2];

    int t = t0;
    for (; t + 1 < t1; t += 2) {
        const float cx0 = x0, cy0 = y0, cz0 = z0;
        {   // unconditional clamped refill: load sinks straight into x0/y0/z0,
            // wait deferred to consumption two tile-bodies later
            const int   tn = (t + 2 < t1) ? (t + 2) : (t1 - 1);
            const size_t o = (size_t)(tn * TILE * 3) + cb;
            x0 = coords[o + 0]; y0 = coords[o + 1]; z0 = coords[o + 2];
        }
        process_tile(t, cx0, cy0, cz0);

        const float cx1 = x1, cy1 = y1, cz1 = z1;
        {
            const int   tn = (t + 3 < t1) ? (t + 3) : (t1 - 1);
            const size_t o = (size_t)(tn * TILE * 3) + cb;
            x1 = coords[o + 0]; y1 = coords[o + 1]; z1 = coords[o + 2];
        }
        process_tile(t + 1, cx1, cy1, cz1);
    }
    if (t < t1) process_tile(t, x0, y0, z0);

    __syncthreads();   // uniform: all waves finished their segments

    // ---- 4-way merge of sorted partial lists -> final ascending top-32 ----
    if (tid < 16) {
        const int r = tid;
        int p0 = TK - 1, p1 = TK - 1, p2 = TK - 1, p3 = TK - 1; // tails = smallest
        for (int k = 0; k < TK; ++k) {
            float d0 = (p0 >= 0) ? s_heap_d[0][r][p0] : 3.0e38f;
            float d1 = (p1 >= 0) ? s_heap_d[1][r][p1] : 3.0e38f;
            float d2 = (p2 >= 0) ? s_heap_d[2][r][p2] : 3.0e38f;
            float d3 = (p3 >= 0) ? s_heap_d[3][r][p3] : 3.0e38f;
            float da = fminf(d0, d1), db = fminf(d2, d3);
            float dm = fminf(da, db);
            int wsel, psel;
            if (dm == d0)      { wsel = 0; psel = p0--; }
            else if (dm == d1) { wsel = 1; psel = p1--; }
            else if (dm == d2) { wsel = 2; psel = p2--; }
            else               { wsel = 3; psel = p3--; }
            s_fin_d[r][k] = dm;
            s_fin_i[r][k] = s_heap_i[wsel][r][psel];
        }
    }
    __syncthreads();

    // ---- Epilogue: COO indices + 64 RBF features per (row,k), 128 threads ----
    for (int e = tid; e < 16 * TK; e += 128) {
        const int  r    = e >> 5;          // row within tile (TK==32)
        const int  k    = e & (TK - 1);
        const int  orow = rt * TILE + r;
        const float d2  = s_fin_d[r][k];   // already ascending
        const int   ci  = s_fin_i[r][k];
        const long  g   = (long)orow * TK + k;
        idx_out[g]                    = orow;  // indices[0][g]
        idx_out[(long)NPTS * TK + g]  = ci;    // indices[1][g]
        const float nd2 = -d2;
        float4* vp4 = (float4*)(val_out + g * NSIG);   // 256B-aligned
        #pragma unroll
        for (int s = 0; s < NSIG; s += 4) {            // 16x global_store_b128
            float4 o;
            o.x = __expf(nd2 * s_inv[s + 0]);
            o.y = __expf(nd2 * s_inv[s + 1]);
            o.z = __expf(nd2 * s_inv[s + 2]);
            o.w = __expf(nd2 * s_inv[s + 3]);
            vp4[s >> 2] = o;
        }
    }
}

extern "C" void kernel_launch(void* const* d_in, const int* in_sizes, int n_in,
                              void* d_out, int out_size, void* d_ws, size_t ws_size,
                              hipStream_t stream) {
    const float* coords = (const float*)d_in[0];   // [12000,3] fp32
    const float* sigma  = (const float*)d_in[1];   // [64] fp32
    // d_out: [2*N*K] int32 indices (bit-stored), then [N*K*64] fp32 values
    int*   idx_out = (int*)d_out;
    float* val_out = (float*)d_out + (size_t)2 * NPTS * TK;
    SparseEdgeEmbedding_knn_rbf<<<NTILES, 128, 0, stream>>>(coords, sigma, idx_out, val_out);
}